// IntSoftmax_38680475468054
// MI455X (gfx1250) — compile-verified
//
#include <hip/hip_runtime.h>
#include <math.h>

// IntSoftmax over rows of 1024 f32. One wave32 per row; 8 rows per 256-thread block.
// gfx1250 paths: clause of global_load_async_to_lds_b128 (ASYNCcnt) with shared
// base + immediate offsets, s_wait_asynccnt, ds_load_b128 consumption,
// non-temporal b128 output stores. VALU-trimmed inner loop: reciprocal-mul
// instead of fp32 divide, v_exp_f32 on the TRANS pipe for 2^(30-q), v_med3 clamp.

#define ROW_LEN        1024
#define WAVES_PER_BLK  8
#define BLK_THREADS    (WAVES_PER_BLK * 32)

typedef float v4f __attribute__((ext_vector_type(4)));

__device__ __forceinline__ float wave_reduce_max(float v) {
#pragma unroll
  for (int off = 16; off > 0; off >>= 1)
    v = fmaxf(v, __shfl_xor(v, off, 32));
  return v;
}

__device__ __forceinline__ float wave_reduce_sum(float v) {
#pragma unroll
  for (int off = 16; off > 0; off >>= 1)
    v += __shfl_xor(v, off, 32);
  return v;
}

__global__ __launch_bounds__(BLK_THREADS)
void intsoftmax_kernel(const float* __restrict__ x,
                       float* __restrict__ out,
                       int nrows) {
  __shared__ __align__(16) float tile[WAVES_PER_BLK][ROW_LEN];

  const int lane = threadIdx.x & 31;
  const int wave = threadIdx.x >> 5;
  const int row  = blockIdx.x * WAVES_PER_BLK + wave;
  if (row >= nrows) return;                      // wave-uniform guard

  const float* rowp = x   + (size_t)row * ROW_LEN;
  float*       outp = out + (size_t)row * ROW_LEN;

  // ---------- async stage: global -> LDS (CDNA5 async-copy engine) ----------
  // INST_OFFSET is added to BOTH the LDS and global addresses (ISA 15.18 async
  // pseudocode), so one global base pair + one LDS-offset VGPR covers all 8
  // transfers via immediate offsets. Low 32 bits of a generic pointer to
  // __shared__ == group-relative LDS byte offset (ISA 10.2).
  {
    const unsigned     loff = (unsigned)(unsigned long long)(void*)&tile[wave][0]
                              + (unsigned)(lane * 16);
    const float* const gp   = rowp + lane * 4;   // 16B-aligned, coalesced
    asm volatile(
        "s_clause 0x7\n\t"
        "global_load_async_to_lds_b128 %0, %1, off\n\t"
        "global_load_async_to_lds_b128 %0, %1, off offset:512\n\t"
        "global_load_async_to_lds_b128 %0, %1, off offset:1024\n\t"
        "global_load_async_to_lds_b128 %0, %1, off offset:1536\n\t"
        "global_load_async_to_lds_b128 %0, %1, off offset:2048\n\t"
        "global_load_async_to_lds_b128 %0, %1, off offset:2560\n\t"
        "global_load_async_to_lds_b128 %0, %1, off offset:3072\n\t"
        "global_load_async_to_lds_b128 %0, %1, off offset:3584"
        :: "v"(loff), "v"(gp) : "memory");
  }
  asm volatile("s_wait_asynccnt 0x0" ::: "memory");  // data visible to this wave

  // ---------- pull row into registers (32 f32/lane), enter x_int domain ------
  float xs[32];
  float m = -INFINITY;
#pragma unroll
  for (int j = 0; j < 8; ++j) {
    v4f v = *(const v4f*)&tile[wave][j * 128 + lane * 4];    // ds_load_b128
#pragma unroll
    for (int k = 0; k < 4; ++k) {
      float xi = v[k] * 1024.0f;                 // x / sf, sf = 2^-10 (exact)
      xs[j * 4 + k] = xi;
      m = fmaxf(m, xi);
    }
  }
  m = wave_reduce_max(m);                        // row max of x_int

  // ---------- constants (all fold to f32 literals; match reference order) ----
  const float x0i    = floorf(-0.6931f * 1024.0f);                    // -710
  const float inv_x0 = 1.0f / x0i;               // reciprocal (folded literal)
  const float bi     = floorf((float)(0.96963238 / 0.35815147) * 1024.0f);
  const float ci     = floorf((float)(1.0 / 0.35815147) * 1048576.0f);
  const float clampv = 30.0f * x0i;                                   // N*x0_int
  // Global max(exp_int) == ci*2^30 exactly (r=0,q=0 at each row's argmax;
  // z<=ci and 2^(30-q)<=2^30 bound all others); global min in [0, max].
  const float act_sf = fmaxf(ci * 0x1p30f, 1e-8f) / 32767.0f;
  const float exp_sf = 0.35815147f * 0x1p-50f;   // (COEF0*sf*sf)/2^30, exact po2
  const float ratio  = exp_sf / act_sf;

  // ---------- int_exp + polynomial + QuantAct requant + row sum --------------
  float lsum = 0.0f;
#pragma unroll
  for (int i = 0; i < 32; ++i) {
    float xi = fmaxf(xs[i] - m, clampv);         // shift + clamp
    float q  = floorf(xi * inv_x0);              // 0..30 (rcp-mul; +/-1ulp safe)
    float r  = fmaf(-x0i, q, xi);                // x_int - x0_int*q (product exact)
    float z  = fmaf(r, r + bi, ci);              // int_polynomial
    float p  = __builtin_amdgcn_exp2f(30.0f - q);// 2^(30-q), TRANS pipe, exact po2
    float e  = floorf(z * p);                    // exp_int (z>0, p>=1 => e>0,
                                                 //  so max(e,0) is an identity)
    float t  = rintf(e * ratio);                 // jnp.round = half-to-even
    t        = __builtin_amdgcn_fmed3f(t, -32768.0f, 32767.0f);  // clip (v_med3)
    xs[i]    = t;                                // exp_int2
    lsum    += t;
  }
  const float s       = wave_reduce_sum(lsum);   // row sum of exp_int2
  const float factor  = floorf(4294967296.0f / s); // 2^32 / sum (inf if sum==0)
  const float factor2 = factor * 0x1p-24f;       // exact po2 scale, commutes w/ RN

  // ---------- normalize + output scale; stream out with NT b128 stores -------
#pragma unroll
  for (int j = 0; j < 8; ++j) {
    v4f o;
#pragma unroll
    for (int k = 0; k < 4; ++k)
      o[k] = floorf(xs[j * 4 + k] * factor2) * 0x1p-8f;
    __builtin_nontemporal_store(o, (v4f*)(outp + j * 128 + lane * 4));
  }
}

extern "C" void kernel_launch(void* const* d_in, const int* in_sizes, int n_in,
                              void* d_out, int out_size, void* d_ws, size_t ws_size,
                              hipStream_t stream) {
  (void)n_in; (void)out_size; (void)d_ws; (void)ws_size;
  const float* x   = (const float*)d_in[0];
  float*       out = (float*)d_out;
  const int n      = in_sizes[0];
  const int nrows  = n / ROW_LEN;                // 4*12*1024 = 49152
  const int blocks = (nrows + WAVES_PER_BLK - 1) / WAVES_PER_BLK;
  intsoftmax_kernel<<<blocks, BLK_THREADS, 0, stream>>>(x, out, nrows);
}